// CMCLLoss_v1_13237089206613
// MI455X (gfx1250) — compile-verified
//
#include <hip/hip_runtime.h>
#include <hip/hip_bf16.h>

// Problem constants (reference: M=4, B=8192, C=1000, BETA=1).
#define MM 4
#define BB 8192
#define CC 1000
#define LOGC 6.907755278982137f        // log(1000)
#define L2E  1.4426950408889634f       // log2(e)
#define LN2  0.6931471805599453f       // ln(2)

// Raw v_exp_f32 (no denorm-range fixup). Safe here: flushed results (< 2^-126)
// are negligible against a running sum-of-exp >= 1.
#if __has_builtin(__builtin_amdgcn_exp2f)
#define EXP2(x) __builtin_amdgcn_exp2f(x)
#else
#define EXP2(x) exp2f(x)
#endif

typedef __attribute__((ext_vector_type(2))) float v2f;
typedef __attribute__((ext_vector_type(8))) float v8f;

// ---------------------------------------------------------------------------
// Kernel A: per-row stats. One wave32 = 16 rows, arranged as the WMMA
// F32_16x16x4 A-matrix: lane m (0..15) streams K={k,k+1}, lane m+16 streams
// K={k+2,k+3} of row (base+m). B = ones(4x16) so D's columns accumulate the
// row sums on the matrix pipe, while the VALU does a branchless online
// max / sum-exp in the exp2 domain on the same registers (EXEC stays all-1s,
// as WMMA requires; no divergence anywhere in the loop).
// ---------------------------------------------------------------------------
__global__ __launch_bounds__(256) void cmcl_stats_kernel(
    const float* __restrict__ logits, const int* __restrict__ targets,
    float* __restrict__ ce, float* __restrict__ ent) {
  const int lane = threadIdx.x & 31;
  const int wave = threadIdx.x >> 5;
  const int mrow = lane & 15;
  const int half = lane >> 4;                     // 0: K={0,1}, 1: K={2,3}
  const int row  = (blockIdx.x * 8 + wave) * 16 + mrow;   // row = m*BB + b
  const float* __restrict__ p = logits + (size_t)row * CC + 2 * half;

  v2f ones; ones[0] = 1.0f; ones[1] = 1.0f;
  v8f acc = {};                                   // running row sums (all cols)
  float my = -__builtin_inff();                   // running max of x*log2(e)
  float se = 0.0f;                                // running sum of 2^(y - my)

  // Software-pipelined: prefetch next chunk while processing current.
  v2f a = *(const v2f*)(p);
  for (int k = 4; k < CC; k += 4) {
    v2f an = *(const v2f*)(p + k);

    float y0 = a[0] * L2E;
    float nm0 = fmaxf(my, y0);
    se = se * EXP2(my - nm0) + EXP2(y0 - nm0);
    float y1 = a[1] * L2E;
    float nm1 = fmaxf(nm0, y1);
    se = se * EXP2(nm0 - nm1) + EXP2(y1 - nm1);
    my = nm1;

    // Matrix-pipe row-sum accumulate: D[m][n] += sum_k A[m][k]*1
    acc = __builtin_amdgcn_wmma_f32_16x16x4_f32(
        false, a, false, ones, (short)0, acc, false, false);
    a = an;
  }
  // Epilogue: last chunk.
  {
    float y0 = a[0] * L2E;
    float nm0 = fmaxf(my, y0);
    se = se * EXP2(my - nm0) + EXP2(y0 - nm0);
    float y1 = a[1] * L2E;
    float nm1 = fmaxf(nm0, y1);
    se = se * EXP2(nm0 - nm1) + EXP2(y1 - nm1);
    my = nm1;
    acc = __builtin_amdgcn_wmma_f32_16x16x4_f32(
        false, a, false, ones, (short)0, acc, false, false);
  }

  // Merge the two half-lane partials (lane m <-> lane m+16) for max / sumexp.
  float mo = __shfl_xor(my, 16, 32);
  float so = __shfl_xor(se, 16, 32);
  float nm = fmaxf(my, mo);
  float S  = se * EXP2(my - nm) + so * EXP2(mo - nm);
  float lse = LN2 * (nm + __log2f(S));            // back to natural-log domain

  // Extract rowsum[mrow] from the D layout: VGPR r, lanes 0-15 -> M=r,
  // lanes 16-31 -> M=r+8. All columns are equal, so grab any column.
  float sel = acc[mrow & 7];
  int srcLane = ((mrow >> 3) << 4) | (mrow & 7);
  float rowsum = __shfl(sel, srcLane, 32);

  if (lane < 16) {
    int b = row & (BB - 1);                       // row = m*BB + b, BB pow2
    int t = targets[b];
    float tl = logits[(size_t)row * CC + t];
    ce[row]  = lse - tl;                          // cross entropy
    ent[row] = lse - rowsum * (1.0f / CC) - LOGC; // -logC - mean(log_softmax)
  }
}

// ---------------------------------------------------------------------------
// Kernel B: per-sample combine + argmin over M=4 (first-min, like jnp.argmin).
// ---------------------------------------------------------------------------
__global__ __launch_bounds__(256) void cmcl_combine_kernel(
    const float* __restrict__ ce, const float* __restrict__ ent,
    float* __restrict__ minIdxOut, float* __restrict__ contrib) {
  int b = blockIdx.x * 256 + threadIdx.x;
  float c[MM], e[MM];
#pragma unroll
  for (int m = 0; m < MM; ++m) {
    c[m] = ce[m * BB + b];
    e[m] = ent[m * BB + b];
  }
  float te = e[0] + e[1] + e[2] + e[3];
  int win = 0;
  float best = c[0] + (te - e[0]);
#pragma unroll
  for (int m = 1; m < MM; ++m) {
    float lm = c[m] + (te - e[m]);
    if (lm < best) { best = lm; win = m; }
  }
  minIdxOut[b] = (float)win;
  // new_loss = sum_b [ (ce[win]-ent[win]) + sum_m ent[m] ] / B
  contrib[b] = (c[win] - e[win] + te) * (1.0f / BB);
}

// ---------------------------------------------------------------------------
// Kernel C: gather oracle_logits[b][:] = logits[min_index[b], b, :].
// Oracle region starts at d_out+1 (4B-aligned) -> scalar copies. The winner
// rows were just streamed by kernel A and fit in the 192MB L2, so this is
// mostly an L2->HBM write.
// ---------------------------------------------------------------------------
__global__ __launch_bounds__(256) void cmcl_gather_kernel(
    const float* __restrict__ logits, const float* __restrict__ minIdx,
    float* __restrict__ oracle) {
  int b = blockIdx.x;
  int mi = (int)minIdx[b];
  const float* __restrict__ src = logits + ((size_t)mi * BB + b) * CC;
  float* __restrict__ dst = oracle + (size_t)b * CC;
  for (int i = threadIdx.x; i < CC; i += 256) dst[i] = src[i];
}

// ---------------------------------------------------------------------------
// Kernel D: deterministic fixed-order reduction of contrib[] -> new_loss.
// ---------------------------------------------------------------------------
__global__ __launch_bounds__(256) void cmcl_finalize_kernel(
    const float* __restrict__ contrib, float* __restrict__ out0) {
  __shared__ float sm[256];
  int t = threadIdx.x;
  float s = 0.0f;
  for (int i = t; i < BB; i += 256) s += contrib[i];
  sm[t] = s;
  __syncthreads();
  for (int k = 128; k > 0; k >>= 1) {
    if (t < k) sm[t] += sm[t + k];
    __syncthreads();
  }
  if (t == 0) out0[0] = sm[0];
}

extern "C" void kernel_launch(void* const* d_in, const int* in_sizes, int n_in,
                              void* d_out, int out_size, void* d_ws, size_t ws_size,
                              hipStream_t stream) {
  (void)in_sizes; (void)n_in; (void)out_size; (void)ws_size;
  const float* logits  = (const float*)d_in[0];
  const int*   targets = (const int*)d_in[1];

  float* out     = (float*)d_out;
  float* newLoss = out;                           // [0]
  float* oracle  = out + 1;                       // [1 .. 1+B*C)
  float* minIdx  = out + 1 + (size_t)BB * CC;     // [1+B*C .. 1+B*C+B)

  float* wsf     = (float*)d_ws;
  float* ce      = wsf;                           // M*B floats
  float* ent     = wsf + MM * BB;                 // M*B floats
  float* contrib = wsf + 2 * MM * BB;             // B floats

  // 32768 rows, 128 rows per 256-thread block (8 waves x 16 rows) -> 256 blocks
  cmcl_stats_kernel<<<(MM * BB) / 128, 256, 0, stream>>>(logits, targets, ce, ent);
  cmcl_combine_kernel<<<BB / 256, 256, 0, stream>>>(ce, ent, minIdx, contrib);
  cmcl_gather_kernel<<<BB, 256, 0, stream>>>(logits, minIdx, oracle);
  cmcl_finalize_kernel<<<1, 256, 0, stream>>>(contrib, newLoss);
}